// NeighborhoodAttention_76261439308266
// MI455X (gfx1250) — compile-verified
//
#include <hip/hip_runtime.h>
#include <math.h>

#define DIMC    128
#define NHEADS  4
#define HDIM    32
#define KW      7
#define QSCALE  0.17677669529663687f   // HDIM^-0.5

#define GEMM_K     128      // both GEMMs have K = 128
#define BLK_M      128      // 8 waves x 16 rows
#define BLK_N      64       // per-block N panel (4 x 16 per wave)
#define LDS_PITCH  80       // 64 data dwords + 16 pad dwords (bank-conflict free)

typedef __attribute__((ext_vector_type(2))) float v2f;
typedef __attribute__((ext_vector_type(8))) float v8f;
typedef unsigned int u32x4 __attribute__((ext_vector_type(4)));
typedef int          i32x4 __attribute__((ext_vector_type(4)));
typedef int          i32x8 __attribute__((ext_vector_type(8)));

// ---------------------------------------------------------------------------
// C = A(MxK) @ B(KxN) + bias. K fixed at 128.
// Workgroup = 8 waves; block tile = 128(M) x 64(N). The 128x64 B panel is
// staged into LDS once via the Tensor Data Mover (TDM), with TDM-inserted
// row padding (pitch 80 dwords) so B-fragment ds reads are bank-conflict
// free (half-waves land on disjoint 16-bank groups). Each wave computes a
// 16x64 tile: one A-fragment (v2f) feeds 4 v_wmma_f32_16x16x4_f32.
//
// Fragment layouts (CDNA5 ISA 7.12.2, 32-bit data):
//   A 16x4 : lanes 0-15 -> M=lane, {v0,v1}={K=0,1}; lanes 16-31 -> {K=2,3}
//   B 4x16 : lanes 0-15 -> N=lane, {v0,v1}={K=0,1}; lanes 16-31 -> {K=2,3}
//   C 16x16: vgpr r, lane l -> M = r + 8*(l>>4), N = l&15
// ---------------------------------------------------------------------------
__global__ __launch_bounds__(256) void wmma_gemm_bias(
    const float* __restrict__ A, const float* __restrict__ B,
    const float* __restrict__ bias, float* __restrict__ C,
    int M, int N, int qscale)
{
    __shared__ float bs[GEMM_K * LDS_PITCH];   // 40 KB

    const int tid  = threadIdx.x;
    const int lane = tid & 31;
    const int wv   = tid >> 5;                 // 0..7
    const int m0   = blockIdx.x * BLK_M + wv * 16;
    const int n0   = blockIdx.y * BLK_N;

    // ---- stage B[0:128, n0:n0+64] into LDS ---------------------------------
#if __has_builtin(__builtin_amdgcn_tensor_load_to_lds) && \
    __has_builtin(__builtin_amdgcn_s_wait_tensorcnt)
    if (wv == 0) {
        const unsigned long long ga = (unsigned long long)(B + n0); // tile start
        const unsigned int lds_b    = (unsigned int)(unsigned long long)(&bs[0]);

        u32x4 g0;
        g0[0] = 1u;                                   // count=1, user mode
        g0[1] = lds_b;                                // lds_addr (bytes)
        g0[2] = (unsigned int)(ga & 0xFFFFFFFFull);   // global_addr[31:0]
        g0[3] = (unsigned int)((ga >> 32) & 0x1FFFFFFull) | (2u << 30); // addr[56:32] | type=2

        i32x8 g1;
        //        data_size=4B  pad_en   pad_interval=64dw  pad_amount=16dw
        g1[0] = (2 << 16)    | (1 << 20) | (5 << 22)     | (15 << 25);
        g1[1] = (N & 0xFFFF) << 16;                   // tensor_dim0[15:0]
        g1[2] = ((N >> 16) & 0xFFFF) | (GEMM_K << 16); // tensor_dim0[31:16] | tensor_dim1[15:0]
        g1[3] = (BLK_N << 16);                        // tensor_dim1[31:16]=0 | tile_dim0=64
        g1[4] = GEMM_K;                               // tile_dim1=128, tile_dim2=0
        g1[5] = N;                                    // tensor_dim0_stride[31:0]
        g1[6] = 0;                                    // stride[47:32]=0, dim1_stride lo=0
        g1[7] = 0;

        i32x4 gz4 = {0, 0, 0, 0};                     // 2-D tensor: groups 2/3 unused
        i32x8 gz8 = {0, 0, 0, 0, 0, 0, 0, 0};
        __builtin_amdgcn_tensor_load_to_lds(g0, g1, gz4, gz4, gz8, 0);
        __builtin_amdgcn_s_wait_tensorcnt(0);
    }
#else
    // fallback: cooperative load with manual padding
    for (int idx = tid; idx < GEMM_K * BLK_N; idx += 256) {
        const int r = idx >> 6, c = idx & 63;
        bs[r * LDS_PITCH + c] = B[(size_t)r * N + n0 + c];
    }
#endif
    __syncthreads();

    // ---- main K loop -------------------------------------------------------
    const int hl   = lane >> 4;          // K-pair selector
    const int ll   = lane & 15;
    const int arow = m0 + ll;

    v8f acc0 = {0,0,0,0,0,0,0,0};
    v8f acc1 = {0,0,0,0,0,0,0,0};
    v8f acc2 = {0,0,0,0,0,0,0,0};
    v8f acc3 = {0,0,0,0,0,0,0,0};

    #pragma unroll 8
    for (int kk = 0; kk < GEMM_K; kk += 4) {
        const int ka = kk + (hl << 1);
        v2f a = *(const v2f*)(A + (size_t)arow * GEMM_K + ka);
        const float* b0 = &bs[ka * LDS_PITCH + ll];
        const float* b1 = &bs[(ka + 1) * LDS_PITCH + ll];
        v2f b;
        b.x = b0[0];  b.y = b1[0];
        acc0 = __builtin_amdgcn_wmma_f32_16x16x4_f32(false, a, false, b, (short)0, acc0, false, false);
        b.x = b0[16]; b.y = b1[16];
        acc1 = __builtin_amdgcn_wmma_f32_16x16x4_f32(false, a, false, b, (short)0, acc1, false, false);
        b.x = b0[32]; b.y = b1[32];
        acc2 = __builtin_amdgcn_wmma_f32_16x16x4_f32(false, a, false, b, (short)0, acc2, false, false);
        b.x = b0[48]; b.y = b1[48];
        acc3 = __builtin_amdgcn_wmma_f32_16x16x4_f32(false, a, false, b, (short)0, acc3, false, false);
    }

    // ---- epilogue: bias (+ q scale), store ---------------------------------
    v8f* accs[4] = {&acc0, &acc1, &acc2, &acc3};
    #pragma unroll
    for (int t = 0; t < 4; ++t) {
        const int bcol = n0 + t * 16 + ll;
        const float bv = bias[bcol];
        const v8f av = *accs[t];
        #pragma unroll
        for (int r = 0; r < 8; ++r) {
            const int row = m0 + r + (hl << 3);
            float v = av[r] + bv;
            if (qscale && bcol < DIMC) v *= QSCALE;
            C[(size_t)row * N + bcol] = v;
        }
    }
}

// ---------------------------------------------------------------------------
// Neighborhood attention core: one wave per (pixel, head); lane = head-dim
// element. 49-neighbor clamped sliding window, online softmax (k/v read
// once), q.k dot via 5-step shfl_xor tree (wave32). All global loads are
// lane-consecutive -> fully coalesced.
// qkv layout: [pix][384], q = cols 0..127 (pre-scaled), k = +128, v = +256.
// ---------------------------------------------------------------------------
__global__ __launch_bounds__(256) void natten_fused(
    const float* __restrict__ qkv, const float* __restrict__ rpb,
    float* __restrict__ attn_out)
{
    const int lane = threadIdx.x & 31;
    const int wid  = blockIdx.x * 8 + (threadIdx.x >> 5);
    const int head = wid & 3;
    const int pix  = wid >> 2;
    const int ii   = (pix >> 6) & 63;
    const int jj   = pix & 63;
    const int bimg = pix >> 12;
    const int si   = min(max(ii - 3, 0), 64 - KW);
    const int sj   = min(max(jj - 3, 0), 64 - KW);

    const int hoff = head * HDIM + lane;
    const float qd = qkv[(size_t)pix * 384 + hoff];

    const float* rpb_h = rpb + head * 13 * 13;

    float mrun = -1e30f, lsum = 0.f, acc = 0.f;

    for (int p = 0; p < KW; ++p) {
        const int bi      = si + p - ii + 6;
        const int rowbase = (bimg << 12) + ((si + p) << 6);
        #pragma unroll
        for (int q = 0; q < KW; ++q) {
            const int np = rowbase + sj + q;
            const float* kv = qkv + (size_t)np * 384 + hoff;
            float s = qd * kv[DIMC];
            s += __shfl_xor(s, 16, 32);
            s += __shfl_xor(s,  8, 32);
            s += __shfl_xor(s,  4, 32);
            s += __shfl_xor(s,  2, 32);
            s += __shfl_xor(s,  1, 32);
            s += rpb_h[bi * 13 + (sj + q - jj + 6)];

            const float vd = kv[2 * DIMC];
            const float mn = fmaxf(mrun, s);
            const float sc = __expf(mrun - mn);
            const float pe = __expf(s - mn);
            lsum = lsum * sc + pe;
            acc  = acc  * sc + pe * vd;
            mrun = mn;
        }
    }
    attn_out[(size_t)pix * DIMC + hoff] = acc / lsum;
}

// ---------------------------------------------------------------------------
extern "C" void kernel_launch(void* const* d_in, const int* in_sizes, int n_in,
                              void* d_out, int out_size, void* d_ws, size_t ws_size,
                              hipStream_t stream) {
    const float* x      = (const float*)d_in[0];
    const float* w_qkv  = (const float*)d_in[1];
    const float* b_qkv  = (const float*)d_in[2];
    const float* rpb    = (const float*)d_in[3];
    const float* w_proj = (const float*)d_in[4];
    const float* b_proj = (const float*)d_in[5];
    float* out = (float*)d_out;

    const int Npix = 4 * 64 * 64;                        // 16384 rows

    float* qkv_ws  = (float*)d_ws;                       // 16384 x 384
    float* attn_ws = qkv_ws + (size_t)Npix * 384;        // 16384 x 128

    // 1) fused QKV projection (q columns scaled by HDIM^-0.5)
    {
        dim3 grid(Npix / BLK_M, 384 / BLK_N);            // 128 x 6
        wmma_gemm_bias<<<grid, 256, 0, stream>>>(
            x, w_qkv, b_qkv, qkv_ws, Npix, 384, 1);
    }

    // 2) 7x7 neighborhood attention, one wave per (pixel, head)
    {
        const int waves = Npix * NHEADS;                 // 65536 waves
        natten_fused<<<waves / 8, 256, 0, stream>>>(qkv_ws, rpb, attn_ws);
    }

    // 3) output projection
    {
        dim3 grid(Npix / BLK_M, 128 / BLK_N);            // 128 x 2
        wmma_gemm_bias<<<grid, 256, 0, stream>>>(
            attn_ws, w_proj, b_proj, out, Npix, 128, 0);
    }
}